// MoELayer_71665824301681
// MI455X (gfx1250) — compile-verified
//
#include <hip/hip_runtime.h>
#include <math.h>

// Problem constants (match reference)
#define cB 4
#define cL 2048
#define cD 1024
#define cH 2048
#define cO 1024
#define cE 8
#define cK 2
#define T  (cB * cL)              // 8192 tokens
#define RMAX (T * cK + cE * 64)   // 16896 padded assignment rows (mult of 64)

typedef float v2f __attribute__((ext_vector_type(2)));
typedef float v8f __attribute__((ext_vector_type(8)));

// ---------------------------------------------------------------------------
// K0: zero routing counters
// ---------------------------------------------------------------------------
__global__ void moe_init(int* __restrict__ counts, int* __restrict__ cursors) {
    if (threadIdx.x < cE) { counts[threadIdx.x] = 0; cursors[threadIdx.x] = 0; }
}

// ---------------------------------------------------------------------------
// K1: gate — one wave per token. logits = x @ Wg + bg ; top-2 ; softmax(2)
// ---------------------------------------------------------------------------
__global__ __launch_bounds__(256) void moe_gate(
    const float* __restrict__ x, const float* __restrict__ Wg,
    const float* __restrict__ bg,
    float* __restrict__ topw, int* __restrict__ topi, int* __restrict__ counts)
{
    int wave = threadIdx.x >> 5;
    int lane = threadIdx.x & 31;
    int t = blockIdx.x * 8 + wave;          // grid = T/8 -> t < T always

    const float* xt = x + (size_t)t * cD;
    float acc[cE];
#pragma unroll
    for (int e = 0; e < cE; ++e) acc[e] = 0.0f;

    for (int d = lane; d < cD; d += 32) {
        float xv = xt[d];
        const float* wr = Wg + (size_t)d * cE;
#pragma unroll
        for (int e = 0; e < cE; ++e) acc[e] += xv * wr[e];
    }
    // wave32 butterfly reduction
#pragma unroll
    for (int e = 0; e < cE; ++e) {
#pragma unroll
        for (int off = 16; off > 0; off >>= 1)
            acc[e] += __shfl_xor(acc[e], off, 32);
    }

    if (lane == 0) {
        float v0 = -INFINITY, v1 = -INFINITY;
        int   i0 = 0,          i1 = 0;
#pragma unroll
        for (int e = 0; e < cE; ++e) {
            float v = acc[e] + bg[e];
            if (v > v0)      { v1 = v0; i1 = i0; v0 = v; i0 = e; }
            else if (v > v1) { v1 = v;  i1 = e; }
        }
        float e1 = expf(v1 - v0);           // softmax over the selected 2
        float s  = 1.0f + e1;
        topw[t * 2 + 0] = 1.0f / s;
        topw[t * 2 + 1] = e1 / s;
        topi[t * 2 + 0] = i0;
        topi[t * 2 + 1] = i1;
        atomicAdd(&counts[i0], 1);
        atomicAdd(&counts[i1], 1);
    }
}

// ---------------------------------------------------------------------------
// K2: exclusive scan of counts, padded to 64-row tiles
// ---------------------------------------------------------------------------
__global__ void moe_offsets(const int* __restrict__ counts, int* __restrict__ offs) {
    if (threadIdx.x == 0) {
        int acc = 0;
        for (int e = 0; e < cE; ++e) {
            offs[e] = acc;
            acc += (counts[e] + 63) & ~63;
        }
        offs[cE] = acc;
    }
}

// ---------------------------------------------------------------------------
// K3: assign each (token,slot) a row in the permuted buffer and gather x
// ---------------------------------------------------------------------------
__global__ __launch_bounds__(256) void moe_gather(
    const float* __restrict__ x, const int* __restrict__ topi,
    const int* __restrict__ offs, int* __restrict__ cursors,
    int* __restrict__ rowOf, float* __restrict__ Xg)
{
    __shared__ int sRow;
    int a = blockIdx.x;                      // assignment index: 0 .. T*cK-1
    if (threadIdx.x == 0) {
        int e = topi[a];
        int r = offs[e] + atomicAdd(&cursors[e], 1);
        rowOf[a] = r;
        sRow = r;
    }
    __syncthreads();
    int t = a >> 1;
    const float4* src = (const float4*)(x  + (size_t)t    * cD);
    float4*       dst = (float4*)      (Xg + (size_t)sRow * cD);
    dst[threadIdx.x] = src[threadIdx.x];     // 256 lanes * float4 = 1024 floats
}

// ---------------------------------------------------------------------------
// K4: expert-segmented GEMM, fp32 WMMA 16x16x4.
//   C[r, n] = act( A[r, :] @ W[e][:, n] + bias[e][n] ),  e from row segment.
//   Block: 256 threads (8 waves), 64x128 C tile, K pipelined in 32-chunks:
//   register prefetch of chunk i+1 overlaps the WMMA work of chunk i, LDS
//   tiles are double buffered (ping-pong).
//   Wave w -> rows (w>>1)*16, cols (w&1)*64 : four 16x16 WMMA accumulators,
//   each A fragment reused 4x.
//   B is stored in LDS k-pair interleaved: sB[k/2][n][k&1] so that a WMMA B
//   fragment {B[k][n], B[k+1][n]} is one contiguous 8B-aligned ds_load_b64
//   (no VGPR repacking movs in the hot loop).
// ---------------------------------------------------------------------------
__global__ __launch_bounds__(256) void moe_gemm(
    const float* __restrict__ A,      // [RMAX, Kd]
    const float* __restrict__ Wb,     // [cE, Kd, Nd]
    const float* __restrict__ bias,   // [cE, Nd]
    float* __restrict__ C,            // [RMAX, Nd]
    const int* __restrict__ offs,     // [cE+1]
    int Kd, int Nd, int doRelu)
{
    __shared__ float sA[2][64 * 36];   // 64 rows x 32 k, stride 36 (conflict-free, 16B aligned)
    __shared__ float sB[2][16 * 256];  // 16 k-pairs x (128 n x 2 k) interleaved

    const int rowBase = blockIdx.x * 64;
    const int nBase   = blockIdx.y * 128;

    if (rowBase >= offs[cE]) return;   // beyond padded assignment rows
    int e = 0;
#pragma unroll
    for (int i = 1; i < cE; ++i) if (rowBase >= offs[i]) e = i;

    const float* W = Wb + (size_t)e * Kd * Nd;

    const int tid    = threadIdx.x;
    const int wave   = tid >> 5;
    const int lane   = tid & 31;
    const int r0     = (wave >> 1) * 16;   // wave's row offset in tile
    const int c0     = (wave & 1) * 64;    // wave's col offset in tile
    const int laneM  = lane & 15;          // M (for A) / N (for B) index
    const int laneHi = lane >> 4;          // 0: K 0..1, 1: K 2..3
    const int koff   = laneHi * 2;

    // staging coordinates (fixed per thread)
    const int arA[2] = { (tid + 0)   >> 3, (tid + 256) >> 3 };   // A row 0..63
    const int acA    = (tid & 7) * 4;                            // A col
    int bpB[4], bnB[4];                    // B: k-pair index, n-pair base
#pragma unroll
    for (int i = 0; i < 4; ++i) {
        int idx = tid + i * 256;           // 0..1023 float4 blocks
        bpB[i] = idx >> 6;                 // k-pair 0..15
        bnB[i] = (idx & 63) * 2;           // n (even) 0..126
    }

    float4 ra[2];
    float2 rb0[4], rb1[4];                 // two adjacent k-rows per block

    auto loadChunk = [&](int k0) {
#pragma unroll
        for (int i = 0; i < 2; ++i)
            ra[i] = *(const float4*)(A + (size_t)(rowBase + arA[i]) * Kd + k0 + acA);
#pragma unroll
        for (int i = 0; i < 4; ++i) {
            const float* wp = W + (size_t)(k0 + bpB[i] * 2) * Nd + nBase + bnB[i];
            rb0[i] = *(const float2*)(wp);
            rb1[i] = *(const float2*)(wp + Nd);
        }
    };
    auto storeChunk = [&](int buf) {
#pragma unroll
        for (int i = 0; i < 2; ++i)
            *(float4*)(&sA[buf][arA[i] * 36 + acA]) = ra[i];
#pragma unroll
        for (int i = 0; i < 4; ++i) {
            float4 v;                      // interleave: {k, k+1, k, k+1}
            v.x = rb0[i].x; v.y = rb1[i].x; v.z = rb0[i].y; v.w = rb1[i].y;
            *(float4*)(&sB[buf][bpB[i] * 256 + bnB[i] * 2]) = v;
        }
    };

    v8f acc[4];
#pragma unroll
    for (int j = 0; j < 4; ++j) acc[j] = (v8f){};

    const int nChunks = Kd >> 5;           // Kd / 32
    loadChunk(0);
    storeChunk(0);
    __syncthreads();

    for (int ch = 0; ch < nChunks; ++ch) {
        const int cur = ch & 1;
        const bool hasNext = (ch + 1) < nChunks;
        if (hasNext) loadChunk((ch + 1) << 5);   // global->reg, overlaps WMMA below

#pragma unroll
        for (int kk = 0; kk < 32; kk += 4) {
            // A fragment 16x4: lane<16 -> K 0..1, lane>=16 -> K 2..3 (VGPR pair)
            v2f a = *(const v2f*)(&sA[cur][(r0 + laneM) * 36 + kk + koff]);
            // B fragments: contiguous k-pair at {n} — single b64 each
            const int pB = (kk + koff) >> 1;     // k-pair index
#pragma unroll
            for (int j = 0; j < 4; ++j) {
                v2f b = *(const v2f*)(&sB[cur][pB * 256 + (c0 + j * 16 + laneM) * 2]);
                acc[j] = __builtin_amdgcn_wmma_f32_16x16x4_f32(
                             false, a, false, b, (short)0, acc[j], false, false);
            }
        }

        if (hasNext) {
            __syncthreads();               // all waves done reading buf (1-cur)
            storeChunk(1 - cur);
            __syncthreads();               // buf (1-cur) ready for next chunk
        }
    }

    // epilogue: bias + optional relu, scatter per C/D layout (VGPR i -> M=i / M=i+8)
    const int mBase = rowBase + r0 + laneHi * 8;
#pragma unroll
    for (int j = 0; j < 4; ++j) {
        const int n0 = nBase + c0 + j * 16 + laneM;
        const float bv = bias[(size_t)e * Nd + n0];
#pragma unroll
        for (int i = 0; i < 8; ++i) {
            float v = acc[j][i] + bv;
            if (doRelu) v = fmaxf(v, 0.0f);
            C[(size_t)(mBase + i) * Nd + n0] = v;
        }
    }
}

// ---------------------------------------------------------------------------
// K5: combine — out[t] = w0 * Y[row0] + w1 * Y[row1]   (no atomics)
// ---------------------------------------------------------------------------
__global__ __launch_bounds__(256) void moe_combine(
    const float* __restrict__ Y, const float* __restrict__ topw,
    const int* __restrict__ rowOf, float* __restrict__ out)
{
    int t  = blockIdx.x;
    int r0 = rowOf[t * 2 + 0];
    int r1 = rowOf[t * 2 + 1];
    float w0 = topw[t * 2 + 0];
    float w1 = topw[t * 2 + 1];
    const float4* y0 = (const float4*)(Y + (size_t)r0 * cO);
    const float4* y1 = (const float4*)(Y + (size_t)r1 * cO);
    float4 a = y0[threadIdx.x];
    float4 b = y1[threadIdx.x];
    float4 o;
    o.x = w0 * a.x + w1 * b.x;
    o.y = w0 * a.y + w1 * b.y;
    o.z = w0 * a.z + w1 * b.z;
    o.w = w0 * a.w + w1 * b.w;
    ((float4*)(out + (size_t)t * cO))[threadIdx.x] = o;
}

// ---------------------------------------------------------------------------
// K6: load-balancing loss. K=2, ddof=1 => var = (u0-u1)^2 / 2 ; mean over B.
// ---------------------------------------------------------------------------
__global__ __launch_bounds__(256) void moe_lbl(
    const float* __restrict__ topw, float* __restrict__ lblOut)
{
    __shared__ float sm[256 * 8];
    float u[8];
#pragma unroll
    for (int j = 0; j < 8; ++j) u[j] = 0.0f;
    for (int l = threadIdx.x; l < cL; l += 256) {
#pragma unroll
        for (int b = 0; b < cB; ++b) {
#pragma unroll
            for (int k = 0; k < cK; ++k)
                u[b * 2 + k] += topw[((size_t)b * cL + l) * 2 + k];
        }
    }
#pragma unroll
    for (int j = 0; j < 8; ++j) sm[threadIdx.x * 8 + j] = u[j];
    __syncthreads();
    if (threadIdx.x == 0) {
        float lbl = 0.0f;
        for (int b = 0; b < cB; ++b) {
            float u0 = 0.0f, u1 = 0.0f;
            for (int i = 0; i < 256; ++i) {
                u0 += sm[i * 8 + b * 2 + 0];
                u1 += sm[i * 8 + b * 2 + 1];
            }
            u0 *= (1.0f / cL);
            u1 *= (1.0f / cL);
            float d = u0 - u1;
            lbl += 0.5f * d * d;            // unbiased var over 2 samples
        }
        *lblOut = lbl * (1.0f / cB);
    }
}

// ---------------------------------------------------------------------------
extern "C" void kernel_launch(void* const* d_in, const int* in_sizes, int n_in,
                              void* d_out, int out_size, void* d_ws, size_t ws_size,
                              hipStream_t stream) {
    const float* x  = (const float*)d_in[0];
    const float* W1 = (const float*)d_in[1];
    const float* b1 = (const float*)d_in[2];
    const float* W2 = (const float*)d_in[3];
    const float* b2 = (const float*)d_in[4];
    const float* W3 = (const float*)d_in[5];
    const float* b3 = (const float*)d_in[6];
    const float* Wg = (const float*)d_in[7];
    const float* bg = (const float*)d_in[8];
    float* out = (float*)d_out;

    // workspace carve-up
    char* ws = (char*)d_ws;
    float* topw   = (float*)ws;  ws += (size_t)T * cK * 4;
    int*   topi   = (int*)ws;    ws += (size_t)T * cK * 4;
    int*   rowOf  = (int*)ws;    ws += (size_t)T * cK * 4;
    int*   counts = (int*)ws;                 // 8
    int*   cursors = counts + cE;             // 8
    int*   offs    = cursors + cE;            // 9
    ws += 256;
    float* Xg = (float*)ws;      ws += (size_t)RMAX * cD * 4;   // reused as Y
    float* H1 = (float*)ws;      ws += (size_t)RMAX * cH * 4;
    float* H2 = (float*)ws;

    moe_init   <<<1, 32, 0, stream>>>(counts, cursors);
    moe_gate   <<<T / 8, 256, 0, stream>>>(x, Wg, bg, topw, topi, counts);
    moe_offsets<<<1, 32, 0, stream>>>(counts, offs);
    moe_gather <<<T * cK, 256, 0, stream>>>(x, topi, offs, cursors, rowOf, Xg);

    dim3 g1(RMAX / 64, cH / 128);
    dim3 g2(RMAX / 64, cH / 128);
    dim3 g3(RMAX / 64, cO / 128);
    moe_gemm<<<g1, 256, 0, stream>>>(Xg, W1, b1, H1, offs, cD, cH, 1);
    moe_gemm<<<g2, 256, 0, stream>>>(H1, W2, b2, H2, offs, cH, cH, 1);
    moe_gemm<<<g3, 256, 0, stream>>>(H2, W3, b3, Xg /*Y*/, offs, cH, cO, 0);

    moe_combine<<<T, 256, 0, stream>>>(Xg, topw, rowOf, out);
    moe_lbl    <<<1, 256, 0, stream>>>(topw, out + (size_t)cB * cL * cO);
}